// GNNModel_69329362092401
// MI455X (gfx1250) — compile-verified
//
#include <hip/hip_runtime.h>

typedef __attribute__((ext_vector_type(2))) float v2f;
typedef __attribute__((ext_vector_type(8))) float v8f;
typedef unsigned int u32x4 __attribute__((ext_vector_type(4)));
typedef int i32x4 __attribute__((ext_vector_type(4)));
typedef int i32x8 __attribute__((ext_vector_type(8)));

#define DIM 128
#define SW_LD 132          // padded LDS row stride (floats) for W tile
#define SH_LD 132          // padded LDS row stride (floats) for h tile
#define ROWS_PER_BLOCK 64  // 4 x 16-row M-tiles per block, B fragments reused 4x

// Tensor Data Mover availability / arity probe (see CDNA5_HIP.md):
//  - builtin exists on both toolchains, but ROCm 7.2 = 5 args,
//    amdgpu-toolchain = 6 args; the TDM header ships only with the latter.
#if defined(__has_builtin)
#if __has_builtin(__builtin_amdgcn_tensor_load_to_lds)
#define USE_TDM 1
#endif
#endif
#ifndef USE_TDM
#define USE_TDM 0
#endif
#if __has_include(<hip/amd_detail/amd_gfx1250_TDM.h>)
#define TDM_6ARG 1
#else
#define TDM_6ARG 0
#endif

#if USE_TDM
// Issue one TDM DMA: load the 128x128 fp32 W matrix into LDS at lds_off,
// padding +4 DWORDs after every 128 DWORDs (one row) -> 132-float LDS rows.
// D# packing per CDNA5 ISA §8.3/§8.4 (group0: count=1/type=2/lds/global addr;
// group1: data_size=4B, pad_enable, pad_interval=128dw, pad_amount=4dw,
// tensor_dim0/1=128, tile_dim0/1=128, stride0=128; groups 2/3 zero for 2-D).
__device__ __forceinline__ void tdm_load_W(const float* W, unsigned lds_off) {
  unsigned long long ga = (unsigned long long)(uintptr_t)W;
  u32x4 g0 = {1u,                                  // count=1 (valid), user mode
              lds_off,                             // lds_addr (bytes)
              (unsigned)(ga & 0xFFFFFFFFu),        // global_addr[31:0]
              (unsigned)(((ga >> 32) & 0x01FFFFFFu) | (2u << 30))}; // addr[56:32] | type=2
  i32x8 g1 = {(int)((2u << 16) |                   // data_size = 4 bytes
                    (1u << 20) |                   // pad_enable
                    (6u << 22) |                   // pad_interval: 128 DWORDs
                    (3u << 25)),                   // pad_amount: 4 DWORDs
              (int)(128u << 16),                   // tensor_dim0 = 128 (bits 79:48, low half)
              (int)(128u << 16),                   // tensor_dim0 hi=0 | tensor_dim1 lo=128
              (int)(128u << 16),                   // tensor_dim1 hi=0 | tile_dim0 = 128
              128,                                 // tile_dim1 = 128, tile_dim2 = 0
              128,                                 // tensor_dim0_stride = 128 elements
              0, 0};                               // stride0 hi / tensor_dim1_stride = 0
  i32x4 gz = {0, 0, 0, 0};
#if TDM_6ARG
  i32x8 gz8 = {0, 0, 0, 0, 0, 0, 0, 0};
  __builtin_amdgcn_tensor_load_to_lds(g0, g1, gz, gz, gz8, 0);
#else
  __builtin_amdgcn_tensor_load_to_lds(g0, g1, gz, gz, 0);
#endif
}
#endif

// ---------------------------------------------------------------------------
// 1) Embedding gather: one wave per node row (128 floats = 32 lanes x float4)
// ---------------------------------------------------------------------------
__global__ __launch_bounds__(256) void embed_kernel(
    const int* __restrict__ x, const float* __restrict__ emb,
    float* __restrict__ h, int n_nodes) {
  int node = blockIdx.x * 8 + (threadIdx.x >> 5);
  if (node >= n_nodes) return;
  int lane = threadIdx.x & 31;
  int tok = x[node];
  const float4* src = (const float4*)(emb + (size_t)tok * DIM);
  float4* dst = (float4*)(h + (size_t)node * DIM);
  dst[lane] = src[lane];
}

// ---------------------------------------------------------------------------
// 2) Degree: init to 1.0 (self-loop), atomic-count edges, then rsqrt in place
// ---------------------------------------------------------------------------
__global__ __launch_bounds__(256) void deg_init_kernel(float* __restrict__ deg, int n) {
  int i = blockIdx.x * 256 + threadIdx.x;
  if (i < n) deg[i] = 1.0f;
}

__global__ __launch_bounds__(256) void deg_count_kernel(
    const int* __restrict__ dst, float* __restrict__ deg, int n_edges) {
  int e = blockIdx.x * 256 + threadIdx.x;
  if (e >= n_edges) return;
  __hip_atomic_fetch_add(&deg[dst[e]], 1.0f, __ATOMIC_RELAXED,
                         __HIP_MEMORY_SCOPE_AGENT);
}

__global__ __launch_bounds__(256) void deg_rsqrt_kernel(float* __restrict__ deg, int n) {
  int i = blockIdx.x * 256 + threadIdx.x;
  if (i < n) deg[i] = rsqrtf(deg[i]);
}

// ---------------------------------------------------------------------------
// 3) GEMM  out[N,128] = h[N,128] @ W[128,128]  via V_WMMA_F32_16X16X4_F32.
//    Block = 256 threads = 8 waves; block owns 64 rows x 128 cols.
//    Wave w owns output cols [16w, 16w+16) and iterates 4 M-tiles, reusing
//    each B fragment for 4 WMMAs. W staged into padded LDS by the Tensor
//    Data Mover (wave 0 issues tensor_load_to_lds, drains TENSORcnt before
//    the barrier); 64-row h tile staged cooperatively by all 8 waves.
//    Fragment layouts per CDNA5 ISA 7.12.2:
//      A 16x4 : lanes 0-15 -> M=lane, {K0,K1}; lanes 16-31 -> M=lane-16, {K2,K3}
//      B 4x16 : lanes 0-15 -> N=lane, rows {K0,K1}; lanes 16-31 -> rows {K2,K3}
//      C/D    : vgpr v: lanes 0-15 M=v, lanes 16-31 M=v+8; N=lane%16
// ---------------------------------------------------------------------------
__global__ __launch_bounds__(256) void gemm_wmma_kernel(
    const float* __restrict__ h, const float* __restrict__ W,
    float* __restrict__ out, int n_rows) {
  __shared__ float sW[DIM * SW_LD];              // ~67.6 KB
  __shared__ float sH[ROWS_PER_BLOCK * SH_LD];   // ~33.8 KB

  int tid  = threadIdx.x;
  int wave = tid >> 5;
  int lane = tid & 31;
  int row0 = blockIdx.x * ROWS_PER_BLOCK;

#if USE_TDM
  // TDM DMA of W (one descriptor per block, issued by wave 0). Flat-LDS
  // aperture: low 32 bits of a generic shared pointer = wave-relative LDS
  // byte offset (ISA §10.2 aperture mapping).
  if (wave == 0) tdm_load_W(W, (unsigned)(uintptr_t)sW);
#else
  // Fallback: cooperative stage of W, 128 rows x 32 float4, padded rows.
  for (int i = tid; i < DIM * (DIM / 4); i += 256) {
    int r = i >> 5;
    int c = i & 31;
    ((float4*)(sW + r * SW_LD))[c] = ((const float4*)(W + r * DIM))[c];
  }
#endif

  // Stage 64-row h tile (zero-fill past n_rows).
  for (int i = tid; i < ROWS_PER_BLOCK * (DIM / 4); i += 256) {
    int r = i >> 5;
    int c = i & 31;
    int gr = row0 + r;
    float4 v = make_float4(0.f, 0.f, 0.f, 0.f);
    if (gr < n_rows) v = ((const float4*)(h + (size_t)gr * DIM))[c];
    ((float4*)(sH + r * SH_LD))[c] = v;
  }

#if USE_TDM
  if (wave == 0) __builtin_amdgcn_s_wait_tensorcnt(0);  // drain TDM before barrier
#endif
  __syncthreads();

  int ncol  = (wave << 4) + (lane & 15);  // output column this lane owns in B/C/D
  int mrow  = lane & 15;                  // A row within each 16-row M-tile
  int khalf = (lane >> 4) << 1;           // 0 for lanes 0-15, 2 for lanes 16-31

  v8f acc0 = {}, acc1 = {}, acc2 = {}, acc3 = {};
#pragma unroll
  for (int k = 0; k < DIM; k += 4) {
    v2f b;
    b.x = sW[(k + khalf + 0) * SW_LD + ncol];
    b.y = sW[(k + khalf + 1) * SW_LD + ncol];
    const float* ap = sH + mrow * SH_LD + k + khalf;
    v2f a0, a1, a2, a3;
    a0.x = ap[0 * 16 * SH_LD + 0];  a0.y = ap[0 * 16 * SH_LD + 1];
    a1.x = ap[1 * 16 * SH_LD + 0];  a1.y = ap[1 * 16 * SH_LD + 1];
    a2.x = ap[2 * 16 * SH_LD + 0];  a2.y = ap[2 * 16 * SH_LD + 1];
    a3.x = ap[3 * 16 * SH_LD + 0];  a3.y = ap[3 * 16 * SH_LD + 1];
    acc0 = __builtin_amdgcn_wmma_f32_16x16x4_f32(false, a0, false, b, (short)0, acc0, false, false);
    acc1 = __builtin_amdgcn_wmma_f32_16x16x4_f32(false, a1, false, b, (short)0, acc1, false, false);
    acc2 = __builtin_amdgcn_wmma_f32_16x16x4_f32(false, a2, false, b, (short)0, acc2, false, false);
    acc3 = __builtin_amdgcn_wmma_f32_16x16x4_f32(false, a3, false, b, (short)0, acc3, false, false);
  }

  int mbase = (lane >> 4) ? 8 : 0;
  v8f accs[4] = {acc0, acc1, acc2, acc3};
#pragma unroll
  for (int t = 0; t < 4; t++) {
#pragma unroll
    for (int v = 0; v < 8; v++) {
      int gr = row0 + t * 16 + mbase + v;
      if (gr < n_rows) out[(size_t)gr * DIM + ncol] = accs[t][v];
    }
  }
}

// ---------------------------------------------------------------------------
// 4) agg[n] = b + dinv[n]^2 * hw[n]   (bias + self-loop message, no atomics)
// ---------------------------------------------------------------------------
__global__ __launch_bounds__(256) void init_self_kernel(
    const float* __restrict__ hw, const float* __restrict__ dinv,
    const float* __restrict__ bias, float* __restrict__ agg, int n_nodes) {
  int node = blockIdx.x * 8 + (threadIdx.x >> 5);
  if (node >= n_nodes) return;
  int lane = threadIdx.x & 31;
  float di = dinv[node];
  float w = di * di;
  float4 v = ((const float4*)(hw + (size_t)node * DIM))[lane];
  float4 b = ((const float4*)bias)[lane];
  float4 o;
  o.x = b.x + w * v.x;
  o.y = b.y + w * v.y;
  o.z = b.z + w * v.z;
  o.w = b.w + w * v.w;
  ((float4*)(agg + (size_t)node * DIM))[lane] = o;
}

// ---------------------------------------------------------------------------
// 5) Edge scatter: one wave per edge; lane gathers float4 of hw[src], scales
//    by dinv[src]*dinv[dst], atomically accumulates into agg[dst] with
//    hardware global_atomic_add_f32 (relaxed / agent scope).
// ---------------------------------------------------------------------------
__global__ __launch_bounds__(256) void edge_scatter_kernel(
    const int* __restrict__ src, const int* __restrict__ dst,
    const float* __restrict__ dinv, const float* __restrict__ hw,
    float* __restrict__ agg, int n_edges) {
  int e = blockIdx.x * 8 + (threadIdx.x >> 5);
  if (e >= n_edges) return;
  int lane = threadIdx.x & 31;
  int s = src[e];
  int t = dst[e];
  __builtin_prefetch(hw + (size_t)s * DIM + lane * 4, 0, 1);  // global_prefetch_b8
  float w = dinv[s] * dinv[t];
  float4 m = ((const float4*)(hw + (size_t)s * DIM))[lane];
  float* out = agg + (size_t)t * DIM + lane * 4;
  __hip_atomic_fetch_add(out + 0, w * m.x, __ATOMIC_RELAXED, __HIP_MEMORY_SCOPE_AGENT);
  __hip_atomic_fetch_add(out + 1, w * m.y, __ATOMIC_RELAXED, __HIP_MEMORY_SCOPE_AGENT);
  __hip_atomic_fetch_add(out + 2, w * m.z, __ATOMIC_RELAXED, __HIP_MEMORY_SCOPE_AGENT);
  __hip_atomic_fetch_add(out + 3, w * m.w, __ATOMIC_RELAXED, __HIP_MEMORY_SCOPE_AGENT);
}

// ---------------------------------------------------------------------------
// 6) ReLU in place over float4 elements
// ---------------------------------------------------------------------------
__global__ __launch_bounds__(256) void relu_kernel(float* __restrict__ a, int n_vec4) {
  int i = blockIdx.x * 256 + threadIdx.x;
  if (i >= n_vec4) return;
  float4 v = ((float4*)a)[i];
  v.x = fmaxf(v.x, 0.f);
  v.y = fmaxf(v.y, 0.f);
  v.z = fmaxf(v.z, 0.f);
  v.w = fmaxf(v.w, 0.f);
  ((float4*)a)[i] = v;
}

// ---------------------------------------------------------------------------
// Launch: embed -> degree -> [gemm -> init(bias+self) -> edge atomics -> relu] x2
// ws layout: h[N*128] | hw[N*128] | dinv[N]   (~103 MB for N=100000)
// ---------------------------------------------------------------------------
extern "C" void kernel_launch(void* const* d_in, const int* in_sizes, int n_in,
                              void* d_out, int out_size, void* d_ws, size_t ws_size,
                              hipStream_t stream) {
  const int*   x    = (const int*)d_in[0];
  const int*   ei   = (const int*)d_in[1];   // [2, E] row-major
  const float* emb  = (const float*)d_in[2];
  const float* W1   = (const float*)d_in[3];
  const float* b1   = (const float*)d_in[4];
  const float* W2   = (const float*)d_in[5];
  const float* b2   = (const float*)d_in[6];
  float* out = (float*)d_out;

  const int N = in_sizes[0];
  const int E = in_sizes[1] / 2;
  const int* src = ei;
  const int* dst = ei + E;

  float* h    = (float*)d_ws;              // [N,128] features / layer-1 agg
  float* hw   = h + (size_t)N * DIM;       // [N,128] h @ W
  float* dinv = hw + (size_t)N * DIM;      // [N]

  const int gNode  = (N + 7) / 8;          // 1 wave / node kernels
  const int gElem  = (N + 255) / 256;
  const int gEdge1 = (E + 255) / 256;
  const int gEdge8 = (E + 7) / 8;
  const int gTile  = (N + ROWS_PER_BLOCK - 1) / ROWS_PER_BLOCK;
  const int gVec4  = (N * (DIM / 4) + 255) / 256;

  // h0 = emb[x]
  embed_kernel<<<gNode, 256, 0, stream>>>(x, emb, h, N);
  // dinv = rsqrt(1 + indegree)
  deg_init_kernel<<<gElem, 256, 0, stream>>>(dinv, N);
  deg_count_kernel<<<gEdge1, 256, 0, stream>>>(dst, dinv, E);
  deg_rsqrt_kernel<<<gElem, 256, 0, stream>>>(dinv, N);

  // ---- layer 1 ----
  gemm_wmma_kernel<<<gTile, 256, 0, stream>>>(h, W1, hw, N);
  init_self_kernel<<<gNode, 256, 0, stream>>>(hw, dinv, b1, h, N);  // h becomes agg
  edge_scatter_kernel<<<gEdge8, 256, 0, stream>>>(src, dst, dinv, hw, h, E);
  relu_kernel<<<gVec4, 256, 0, stream>>>(h, N * (DIM / 4));

  // ---- layer 2 (aggregate straight into d_out) ----
  gemm_wmma_kernel<<<gTile, 256, 0, stream>>>(h, W2, hw, N);
  init_self_kernel<<<gNode, 256, 0, stream>>>(hw, dinv, b2, out, N);
  edge_scatter_kernel<<<gEdge8, 256, 0, stream>>>(src, dst, dinv, hw, out, E);
  relu_kernel<<<gVec4, 256, 0, stream>>>(out, N * (DIM / 4));
}